// Social_Aggregator_74431783239690
// MI455X (gfx1250) — compile-verified
//
#include <hip/hip_runtime.h>

// CDNA5 / gfx1250: wave32, WMMA (f32 16x16x4 keeps full precision).
typedef float v2f __attribute__((ext_vector_type(2)));
typedef float v8f __attribute__((ext_vector_type(8)));

#define EMBED_DIM       64
#define MAX_NB          50
#define K_PAD           52      // pad 50 -> 52 (multiple of 4), pad weights = 0
#define WAVES_PER_BLOCK 8
#define BLOCK_THREADS   (WAVES_PER_BLOCK * 32)

__global__ __launch_bounds__(BLOCK_THREADS)
void social_agg_wmma_f32(const int* __restrict__ nodes,
                         const int* __restrict__ u_u,      // [N_USERS, 50]
                         const float* __restrict__ u_u_l,  // [N_USERS]
                         const float* __restrict__ emb,    // [N_USERS, 64]
                         float* __restrict__ out,          // [B, 64]
                         int batch) {
    __shared__ float s_w[WAVES_PER_BLOCK][K_PAD];
    __shared__ int   s_nb[WAVES_PER_BLOCK][K_PAD];

    const int wave = threadIdx.x >> 5;
    const int lane = threadIdx.x & 31;
    const int b_raw = blockIdx.x * WAVES_PER_BLOCK + wave;
    const int b = (b_raw < batch) ? b_raw : 0;   // clamp; store predicated below

    const int   node = nodes[b];
    const float Na   = u_u_l[node];
    const float sNa  = sqrtf(Na);                // exact-path 1/(sqrt*sqrt) like reference

    // ---- gather neighbor ids + degrees, build weights in LDS -------------
    const int base = node * MAX_NB;

    // k = lane (0..31)
    int   nb0 = u_u[base + lane];
    float Nb0 = u_u_l[nb0];
    float w0  = (Na > 0.0f && Nb0 > 0.0f) ? 1.0f / (sNa * sqrtf(Nb0)) : 0.0f;
    s_nb[wave][lane] = nb0;
    s_w[wave][lane]  = w0;

    // prefetch this lane's embedding row (256B = 2 lines) -> global_prefetch_b8
    {
        const char* p = (const char*)(emb + (size_t)nb0 * EMBED_DIM);
        __builtin_prefetch(p, 0, 3);
        __builtin_prefetch(p + 128, 0, 3);
    }

    // k = 32 + lane (32..51): real for k<50, zero-weight padding for 50,51
    if (lane < (K_PAD - 32)) {
        const int k   = 32 + lane;
        int   nb1 = 0;
        float w1  = 0.0f;
        if (k < MAX_NB) {
            nb1 = u_u[base + k];
            float Nb1 = u_u_l[nb1];
            w1 = (Na > 0.0f && Nb1 > 0.0f) ? 1.0f / (sNa * sqrtf(Nb1)) : 0.0f;
            const char* p = (const char*)(emb + (size_t)nb1 * EMBED_DIM);
            __builtin_prefetch(p, 0, 3);
            __builtin_prefetch(p + 128, 0, 3);
        }
        s_nb[wave][k] = nb1;
        s_w[wave][k]  = w1;
    }
    __syncthreads();   // all threads reach this (no early returns)

    // ---- weighted reduction on the matrix pipe ---------------------------
    // D(16x16) += A(16x4: w broadcast over M) x B(4x16: E[nb[k], d-chunk])
    // A/B f32 layout: lanes 0-15 hold K = k0,k0+1 (VGPR0,VGPR1),
    //                 lanes 16-31 hold K = k0+2,k0+3.
    const int half = lane >> 4;        // 0 | 1
    const int nsub = lane & 15;        // N column within tile = d within chunk

    v8f acc0 = {}; v8f acc1 = {}; v8f acc2 = {}; v8f acc3 = {};

#pragma unroll
    for (int k0 = 0; k0 < K_PAD; k0 += 4) {
        const int idx = k0 + (half << 1);
        v2f a;
        a.x = s_w[wave][idx];
        a.y = s_w[wave][idx + 1];

        const size_t r0 = (size_t)s_nb[wave][idx]     * EMBED_DIM;
        const size_t r1 = (size_t)s_nb[wave][idx + 1] * EMBED_DIM;

        v2f b0, b1, b2, b3;
        b0.x = emb[r0 + 0  + nsub];  b0.y = emb[r1 + 0  + nsub];
        b1.x = emb[r0 + 16 + nsub];  b1.y = emb[r1 + 16 + nsub];
        b2.x = emb[r0 + 32 + nsub];  b2.y = emb[r1 + 32 + nsub];
        b3.x = emb[r0 + 48 + nsub];  b3.y = emb[r1 + 48 + nsub];

        acc0 = __builtin_amdgcn_wmma_f32_16x16x4_f32(false, a, false, b0,
                                                     (short)0, acc0, false, false);
        acc1 = __builtin_amdgcn_wmma_f32_16x16x4_f32(false, a, false, b1,
                                                     (short)0, acc1, false, false);
        acc2 = __builtin_amdgcn_wmma_f32_16x16x4_f32(false, a, false, b2,
                                                     (short)0, acc2, false, false);
        acc3 = __builtin_amdgcn_wmma_f32_16x16x4_f32(false, a, false, b3,
                                                     (short)0, acc3, false, false);
    }

    // D row M=0 lives in VGPR0 (element 0) of lanes 0..15; every row is
    // identical (A rows broadcast), so lanes 0-15 hold out[b, chunk*16 + n].
    if (b_raw < batch && lane < 16) {
        float* o = out + (size_t)b_raw * EMBED_DIM + nsub;
        o[0]  = acc0[0];
        o[16] = acc1[0];
        o[32] = acc2[0];
        o[48] = acc3[0];
    }
}

extern "C" void kernel_launch(void* const* d_in, const int* in_sizes, int n_in,
                              void* d_out, int out_size, void* d_ws, size_t ws_size,
                              hipStream_t stream) {
    const int*   nodes = (const int*)d_in[0];    // [B] int32
    const int*   u_u   = (const int*)d_in[1];    // [N_USERS*50] int32
    const float* u_u_l = (const float*)d_in[2];  // [N_USERS] f32
    const float* emb   = (const float*)d_in[3];  // [N_USERS*64] f32
    float*       out   = (float*)d_out;          // [B*64] f32

    const int batch  = in_sizes[0];
    const int blocks = (batch + WAVES_PER_BLOCK - 1) / WAVES_PER_BLOCK;

    social_agg_wmma_f32<<<blocks, BLOCK_THREADS, 0, stream>>>(
        nodes, u_u, u_u_l, emb, out, batch);
}